// ContrastiveModel_592705487424
// MI455X (gfx1250) — compile-verified
//
#include <hip/hip_runtime.h>
#include <math.h>

typedef __attribute__((ext_vector_type(2))) float v2f;
typedef __attribute__((ext_vector_type(4))) float v4f;
typedef __attribute__((ext_vector_type(8))) float v8f;

#define B_    8
#define D_    32
#define H_    48
#define W_    48
#define HW_   (H_ * W_)          // 2304
#define M_    (B_ * HW_)         // 18432
#define K_    8192
#define NNEG  5
#define NCOL  (B_ + K_)          // 8200
#define TINV  2.0f               // 1/T, T = 0.5

// d_out layout (flat float32, reference tuple order)
#define OFF_TMP   ((size_t)M_ * NCOL)                 // 151142400
#define OFF_LLOG  (OFF_TMP + (size_t)M_)              // + 18432
#define OFF_LLAB  (OFF_LLOG + (size_t)M_ * (NNEG+1))  // + 110592
#define OFF_SLOSS (OFF_LLAB + (size_t)M_)             // + 18432

// LDS geometry for the GEMM (floats)
#define APAD  36     // 64 x 36  (stride pad: conflict-free b64 frag reads, 16B rows)
#define BPAD  36     // 128 x 36
#define CPAD  132    // 64 x 132 (528B row stride: 16B-aligned b128, conflict-free writes)
#define SMEM_A_OFF 0
#define SMEM_B_OFF (64 * APAD)                  // 2304
#define SMEM_FLOATS (64 * CPAD)                 // 8448 floats = 33 KB (C reuses A|B region)

// ---------------------------------------------------------------------------
// gfx1250 async global->LDS staging (ASYNCcnt path), with safe fallback.
// Probe result: builtin exists and takes generic pointers to int32x4
// (vector_size(16)), signature (gsrc, lds_dst, imm offset, imm cpol).
// ---------------------------------------------------------------------------
#if defined(__has_builtin)
#if __has_builtin(__builtin_amdgcn_global_load_async_to_lds_b128)
#define USE_ASYNC_LDS 1
#endif
#endif

typedef int v4i_vs __attribute__((vector_size(4 * sizeof(int))));

__device__ __forceinline__ void stage_b128(float* lds_dst, const float* g_src) {
#if defined(USE_ASYNC_LDS)
  __builtin_amdgcn_global_load_async_to_lds_b128((v4i_vs*)g_src, (v4i_vs*)lds_dst,
                                                 /*imm offset*/0, /*cpol*/0);
#else
  *(v4f*)lds_dst = *(const v4f*)g_src;
#endif
}

__device__ __forceinline__ void stage_wait() {
#if defined(USE_ASYNC_LDS)
#if __has_builtin(__builtin_amdgcn_s_wait_asynccnt)
  __builtin_amdgcn_s_wait_asynccnt(0);
#else
  asm volatile("s_wait_asynccnt 0x0" ::: "memory");
#endif
#endif
}

// ---------------------------------------------------------------------------
// Prep 1: qn = normalize(q_feat, axis=D); emit row-major (M, 32) = WMMA A panel
// ---------------------------------------------------------------------------
__global__ __launch_bounds__(256) void qnorm_kernel(const float* __restrict__ qfeat,
                                                    float* __restrict__ qf) {
  int i = blockIdx.x * blockDim.x + threadIdx.x;
  if (i >= M_) return;
  int b = i / HW_, hw = i % HW_;
  const float* src = qfeat + (size_t)b * D_ * HW_ + hw;
  float v[D_]; float ss = 0.f;
#pragma unroll
  for (int d = 0; d < D_; ++d) { v[d] = src[(size_t)d * HW_]; ss += v[d] * v[d]; }
  float inv = 1.0f / fmaxf(sqrtf(ss), 1e-12f);
  float* dst = qf + (size_t)i * D_;
#pragma unroll
  for (int d = 0; d < D_; ++d) dst[d] = v[d] * inv;
}

// ---------------------------------------------------------------------------
// Prep 2: queue_n = normalize(queue, axis=0), stored TRANSPOSED (K_, 32) so the
// WMMA B fragment is a contiguous 8-byte load per lane (same layout as A).
// ---------------------------------------------------------------------------
__global__ __launch_bounds__(256) void queuent_kernel(const float* __restrict__ queue,
                                                      float* __restrict__ Bt) {
  int n = blockIdx.x * blockDim.x + threadIdx.x;
  if (n >= K_) return;
  float v[D_]; float ss = 0.f;
#pragma unroll
  for (int k = 0; k < D_; ++k) { v[k] = queue[(size_t)k * K_ + n]; ss += v[k] * v[k]; }
  float inv = 1.0f / fmaxf(sqrtf(ss), 1e-12f);
  float* dst = Bt + (size_t)n * D_;
#pragma unroll
  for (int k = 0; k < D_; ++k) dst[k] = v[k] * inv;
}

// ---------------------------------------------------------------------------
// Prep 3: protos[b,:] = normalize( sum_hw normalize(k_feat[b,:,hw]) * sal_k[b,hw] )
// One workgroup per batch; LDS float atomics (ds_add_f32) for the reduction.
// ---------------------------------------------------------------------------
__global__ __launch_bounds__(256) void protos_kernel(const float* __restrict__ kfeat,
                                                     const float* __restrict__ salk,
                                                     float* __restrict__ protosN) {
  __shared__ float sacc[D_ + 1];
  int b = blockIdx.x;
  if (threadIdx.x < D_ + 1) sacc[threadIdx.x] = 0.f;
  __syncthreads();

  float acc[D_];
#pragma unroll
  for (int d = 0; d < D_; ++d) acc[d] = 0.f;

  const float* kb = kfeat + (size_t)b * D_ * HW_;
  const float* sb = salk + (size_t)b * HW_;
  for (int hw = threadIdx.x; hw < HW_; hw += blockDim.x) {
    float v[D_]; float ss = 0.f;
#pragma unroll
    for (int d = 0; d < D_; ++d) { v[d] = kb[(size_t)d * HW_ + hw]; ss += v[d] * v[d]; }
    float s = sb[hw] / fmaxf(sqrtf(ss), 1e-12f);
#pragma unroll
    for (int d = 0; d < D_; ++d) acc[d] += v[d] * s;
  }
#pragma unroll
  for (int d = 0; d < D_; ++d) atomicAdd(&sacc[d], acc[d]);
  __syncthreads();
  if (threadIdx.x == 0) {
    float ss = 0.f;
    for (int d = 0; d < D_; ++d) ss += sacc[d] * sacc[d];
    sacc[D_] = 1.0f / fmaxf(sqrtf(ss), 1e-12f);
  }
  __syncthreads();
  if (threadIdx.x < D_) protosN[b * D_ + threadIdx.x] = sacc[threadIdx.x] * sacc[D_];
}

// ---------------------------------------------------------------------------
// Main GEMM: logits[:, 8:8200] = (q_flat @ queue_n) / T via V_WMMA_F32_16X16X4_F32.
//
// WG tile 64M x 128N (8 waves: 4 along M, 2 along N); wave tile 16M x 64N
// (4 accumulators sharing one A fragment).
//  Phase 1: async-stage A panel (64x32, contiguous 8KB) and B panel (128x32,
//           contiguous 16KB) into padded LDS  -> L2 loads 24KB/WG vs 80KB before.
//  Phase 2: fragment ds_load_b64 reads (conflict-free, stride-36 pad) + 32 WMMAs.
//  Phase 3: repack C through LDS (stride-132 pad) and emit b128 NT stores:
//           each store instruction covers 512B contiguous of one output row.
// ---------------------------------------------------------------------------
__global__ __launch_bounds__(256) void gemm_mem_kernel(const float* __restrict__ qf,
                                                       const float* __restrict__ Bt,
                                                       float* __restrict__ out) {
  __shared__ float smem[SMEM_FLOATS];

  const int tid  = threadIdx.x;
  const int lane = tid & 31;
  const int wave = tid >> 5;
  const int wm = wave & 3;       // M sub-tile (0..3)
  const int wn = wave >> 2;      // N sub-tile (0..1)
  const int ml = lane & 15;
  const int kh = lane >> 4;
  const int m0wg = blockIdx.y * 64;
  const int n0wg = blockIdx.x * 128;

  float* smemA = smem + SMEM_A_OFF;
  float* smemB = smem + SMEM_B_OFF;

  // ---- Phase 1: stage A (2048 floats) and B (4096 floats) panels ----
  const float* gA = qf + (size_t)m0wg * D_;
  const float* gB = Bt + (size_t)n0wg * D_;
#pragma unroll
  for (int i = 0; i < 2; ++i) {
    int e = i * 256 + tid;                 // 0..511 b128 chunks
    int r = e >> 3, c4 = (e & 7) << 2;
    stage_b128(&smemA[r * APAD + c4], gA + r * D_ + c4);
  }
#pragma unroll
  for (int i = 0; i < 4; ++i) {
    int e = i * 256 + tid;                 // 0..1023 b128 chunks
    int r = e >> 3, c4 = (e & 7) << 2;
    stage_b128(&smemB[r * BPAD + c4], gB + r * D_ + c4);
  }
  stage_wait();
  __syncthreads();

  // ---- Phase 2: WMMA f32 16x16x4, K = 32 in 8 chunks ----
  const float* la = smemA + (wm * 16 + ml) * APAD + kh * 2;
  const float* lb = smemB + (wn * 64 + ml) * BPAD + kh * 2;

  v8f acc0 = {}, acc1 = {}, acc2 = {}, acc3 = {};
#pragma unroll
  for (int kc = 0; kc < 8; ++kc) {
    v2f a  = *(const v2f*)(la + kc * 4);
    v2f b0 = *(const v2f*)(lb + kc * 4);
    v2f b1 = *(const v2f*)(lb + 16 * BPAD + kc * 4);
    v2f b2 = *(const v2f*)(lb + 32 * BPAD + kc * 4);
    v2f b3 = *(const v2f*)(lb + 48 * BPAD + kc * 4);
    acc0 = __builtin_amdgcn_wmma_f32_16x16x4_f32(false, a, false, b0, (short)0, acc0, false, false);
    acc1 = __builtin_amdgcn_wmma_f32_16x16x4_f32(false, a, false, b1, (short)0, acc1, false, false);
    acc2 = __builtin_amdgcn_wmma_f32_16x16x4_f32(false, a, false, b2, (short)0, acc2, false, false);
    acc3 = __builtin_amdgcn_wmma_f32_16x16x4_f32(false, a, false, b3, (short)0, acc3, false, false);
  }
  __syncthreads();   // all fragment reads done; smem can be reused for C

  // ---- Phase 3a: accumulators -> LDS C tile (64 x 132) ----
  {
    float* smemC = smem;
    int rbase = wm * 16 + kh * 8;
    int cbase = wn * 64 + ml;
#pragma unroll
    for (int r = 0; r < 8; ++r) {
      float* row = smemC + (size_t)(rbase + r) * CPAD + cbase;
      row[0]  = acc0[r] * TINV;
      row[16] = acc1[r] * TINV;
      row[32] = acc2[r] * TINV;
      row[48] = acc3[r] * TINV;
    }
  }
  __syncthreads();

  // ---- Phase 3b: b128 non-temporal streaming stores (512B/row-segment/wave) ----
  {
    const float* smemC = smem;
#pragma unroll
    for (int i = 0; i < 8; ++i) {
      int e = i * 256 + tid;               // 0..2047
      int rr  = e >> 5;                    // 0..63  (constant per wave per iter)
      int cc4 = (e & 31) << 2;             // 0..124 (lane*4)
      v4f v = *(const v4f*)&smemC[rr * CPAD + cc4];
      size_t base = (size_t)(m0wg + rr) * NCOL + B_ + n0wg + cc4;  // 16B aligned
      __builtin_nontemporal_store(v, (v4f*)(out + base));
    }
  }
}

// ---------------------------------------------------------------------------
// logits[:, 0:8] = (q_flat @ protos.T) / T  (tiny N=8 block -> plain VALU)
// ---------------------------------------------------------------------------
__global__ __launch_bounds__(256) void lbatch_kernel(const float* __restrict__ qf,
                                                     const float* __restrict__ protosN,
                                                     float* __restrict__ out) {
  int idx = blockIdx.x * blockDim.x + threadIdx.x;
  if (idx >= M_ * B_) return;
  int m = idx >> 3, b = idx & 7;
  const float* q = qf + (size_t)m * D_;
  const float* p = protosN + b * D_;
  float s = 0.f;
#pragma unroll
  for (int d = 0; d < D_; ++d) s += q[d] * p[d];
  out[(size_t)m * NCOL + b] = s * TINV;
}

// ---------------------------------------------------------------------------
// tmp[i] = floor(((sal_q[i] + 2*b) * sal_q[i]) / 2),  b = i / HW
// ---------------------------------------------------------------------------
__global__ __launch_bounds__(256) void tmp_kernel(const float* __restrict__ salq,
                                                  float* __restrict__ out) {
  int i = blockIdx.x * blockDim.x + threadIdx.x;
  if (i >= M_) return;
  int b = i / HW_;
  float sq = salq[i];
  out[OFF_TMP + i] = floorf((sq + 2.0f * (float)b) * sq * 0.5f);
}

// ---------------------------------------------------------------------------
// l_logits: positive = <q_i, q_neighbor>, negatives = <q_i, q_rand>.
// JAX threefry gumbel draws are not reproducible device-side; a deterministic
// integer hash stands in for direction / negative-index sampling.
// ---------------------------------------------------------------------------
__device__ __forceinline__ unsigned hash_u32(unsigned x) {
  x ^= x >> 16; x *= 0x7feb352du; x ^= x >> 15; x *= 0x846ca68bu; x ^= x >> 16;
  return x;
}

__global__ __launch_bounds__(256) void llogits_kernel(const float* __restrict__ qf,
                                                      float* __restrict__ out) {
  int i = blockIdx.x * blockDim.x + threadIdx.x;
  if (i >= M_) return;
  int b = i / HW_, hw = i % HW_;
  int r = hw / W_, c = hw % W_;
  unsigned u = hash_u32((unsigned)i * 2654435761u + 0x9e3779b9u);

  const int dr[4] = {1, -1, 0, 0};
  const int dc[4] = {0, 0, 1, -1};
  int cand[4]; int ncand = 0;
#pragma unroll
  for (int j = 0; j < 4; ++j) {
    int rr = r + dr[j], cc = c + dc[j];
    if (rr >= 0 && rr < H_ && cc >= 0 && cc < W_) cand[ncand++] = rr * W_ + cc;
  }
  int pos = cand[u % (unsigned)ncand];

  const float* q  = qf + (size_t)i * D_;
  const float* qb = qf + (size_t)b * HW_ * D_;

  const float* p = qb + (size_t)pos * D_;
  float dp = 0.f;
#pragma unroll
  for (int d = 0; d < D_; ++d) dp += q[d] * p[d];
  out[OFF_LLOG + (size_t)i * (NNEG + 1)] = dp * TINV;

  for (int k = 0; k < NNEG; ++k) {
    unsigned v = hash_u32(u + 0x85ebca6bu * (unsigned)(k + 1)) % (unsigned)HW_;
    if ((int)v == hw) v = (v + 1u) % (unsigned)HW_;
    const float* nptr = qb + (size_t)v * D_;
    float dn = 0.f;
#pragma unroll
    for (int d = 0; d < D_; ++d) dn += q[d] * nptr[d];
    out[OFF_LLOG + (size_t)i * (NNEG + 1) + 1 + k] = dn * TINV;
  }
}

// l_labels = zeros
__global__ __launch_bounds__(256) void labels_kernel(float* __restrict__ out) {
  int i = blockIdx.x * blockDim.x + threadIdx.x;
  if (i < M_) out[OFF_LLAB + i] = 0.f;
}

// ---------------------------------------------------------------------------
// sal_loss: balanced BCE reduction, single workgroup (18432 elements).
// ---------------------------------------------------------------------------
__global__ __launch_bounds__(256) void salloss_kernel(const float* __restrict__ qbg,
                                                      const float* __restrict__ salq,
                                                      float* __restrict__ out) {
  __shared__ float s_np[256], s_lp[256], s_ln[256];
  float np = 0.f, lp = 0.f, ln_ = 0.f;
  for (int i = threadIdx.x; i < M_; i += 256) {
    float x = qbg[i], y = salq[i];
    float t = log1pf(expf(-fabsf(x)));
    float sp_neg = fmaxf(-x, 0.f) + t;   // softplus(-x)
    float sp_pos = fmaxf(x, 0.f) + t;    // softplus(x)
    float loss = sp_neg * y + sp_pos * (1.f - y);
    np += y; lp += y * loss; ln_ += (1.f - y) * loss;
  }
  int t = threadIdx.x;
  s_np[t] = np; s_lp[t] = lp; s_ln[t] = ln_;
  __syncthreads();
  for (int off = 128; off > 0; off >>= 1) {
    if (t < off) { s_np[t] += s_np[t + off]; s_lp[t] += s_lp[t + off]; s_ln[t] += s_ln[t + off]; }
    __syncthreads();
  }
  if (t == 0) {
    float numel = (float)M_;
    float npos = s_np[0];
    float nneg = numel - npos;
    out[OFF_SLOSS] = (nneg / numel * s_lp[0] + npos / numel * s_ln[0]) / numel;
  }
}

// ---------------------------------------------------------------------------
extern "C" void kernel_launch(void* const* d_in, const int* in_sizes, int n_in,
                              void* d_out, int out_size, void* d_ws, size_t ws_size,
                              hipStream_t stream) {
  const float* q_feat = (const float*)d_in[0];
  const float* k_feat = (const float*)d_in[1];
  const float* q_bg   = (const float*)d_in[2];
  const float* sal_q  = (const float*)d_in[3];
  const float* sal_k  = (const float*)d_in[4];
  const float* queue  = (const float*)d_in[5];
  float* out = (float*)d_out;

  float* qf      = (float*)d_ws;                 // (M_, 32)   2.25 MB
  float* Bt      = qf + (size_t)M_ * D_;         // (K_, 32)   1.00 MB
  float* protosN = Bt + (size_t)K_ * D_;         // (8, 32)

  qnorm_kernel  <<<(M_ + 255) / 256, 256, 0, stream>>>(q_feat, qf);
  queuent_kernel<<<(K_ + 255) / 256, 256, 0, stream>>>(queue, Bt);
  protos_kernel <<<B_, 256, 0, stream>>>(k_feat, sal_k, protosN);

  dim3 grid(K_ / 128, M_ / 64);                  // (64, 288)
  gemm_mem_kernel<<<grid, 256, 0, stream>>>(qf, Bt, out);

  lbatch_kernel <<<(M_ * B_ + 255) / 256, 256, 0, stream>>>(qf, protosN, out);
  tmp_kernel    <<<(M_ + 255) / 256, 256, 0, stream>>>(sal_q, out);
  llogits_kernel<<<(M_ + 255) / 256, 256, 0, stream>>>(qf, out);
  labels_kernel <<<(M_ + 255) / 256, 256, 0, stream>>>(out);
  salloss_kernel<<<1, 256, 0, stream>>>(q_bg, sal_q, out);
}